// Filter_Back_Projection_23321672417442
// MI455X (gfx1250) — compile-verified
//
#include <hip/hip_runtime.h>
#include <hip/hip_bf16.h>
#include <math.h>

// ---------------------------------------------------------------------------
// FBP for sino (B=2, A=181, L=512), angles (A,), circle flag.
//  P = 1024. Filtering == circular convolution with g = Re(ifft(filt)),
//  realized as a (B*A x 512) x (512 x 512) f32 GEMM on v_wmma_f32_16x16x4_f32.
//  Backprojection: 1 thread / pixel, LDS-cached cos/sin tables.
// ---------------------------------------------------------------------------

typedef float v2f __attribute__((ext_vector_type(2)));
typedef float v8f __attribute__((ext_vector_type(8)));

#define LP 512
#define PP 1024
#define PI_D 3.14159265358979323846

// ---- Kernel 1: filt[k] = ramp[k] * fftshift(hanning(P))[k]  (f64) ----------
__global__ void __launch_bounds__(256) fbp_build_filt(double* __restrict__ filt) {
    int k = blockIdx.x * 256 + threadIdx.x;           // 0..1023
    // ramp[k] = 2 * sum_m f[m] * cos(2*pi*k*m/P); f[0]=0.25, odd m only else.
    double s = 0.25;                                   // m = 0 term
    for (int j = 0; j < PP / 2; ++j) {
        int    m  = 2 * j + 1;
        double n  = (j < PP / 4) ? (double)(2 * j + 1) : (double)(PP - 1 - 2 * j);
        double fm = -1.0 / ((PI_D * n) * (PI_D * n));
        s += fm * cos(2.0 * PI_D * (double)k * (double)m / (double)PP);
    }
    double ramp = 2.0 * s;
    int    kw   = (k + PP / 2) & (PP - 1);             // fftshift
    double w    = 0.5 - 0.5 * cos(2.0 * PI_D * (double)kw / (double)(PP - 1));
    filt[k] = ramp * w;
}

// ---- Kernel 2: g[n] = (1/P) * sum_k filt[k] * cos(2*pi*k*n/P) --------------
__global__ void __launch_bounds__(256) fbp_build_g(const double* __restrict__ filt,
                                                   float* __restrict__ g) {
    int n = blockIdx.x * 256 + threadIdx.x;            // 0..1023
    double s = 0.0;
    for (int k = 0; k < PP; ++k)
        s += filt[k] * cos(2.0 * PI_D * (double)k * (double)n / (double)PP);
    g[n] = (float)(s / (double)PP);
}

// ---- Kernel 3: filtered sinogram GEMM on f32 WMMA --------------------------
// sf[r, l] = sum_{k<512} sino[r, k] * g[(l - k) mod 1024]
// Block = 128 threads (4 waves). All 4 waves share one 16-row A tile (LDS);
// each wave owns one 16-col N tile. K consumed 4 at a time by WMMA.
__global__ void __launch_bounds__(128) fbp_filter_gemm(const float* __restrict__ sino,
                                                       const float* __restrict__ g,
                                                       float* __restrict__ sf, int R) {
    __shared__ float ldsA[16 * 516];                   // pitch 516 -> conflict-free
    __shared__ float ldsG[PP];

    const int tid   = threadIdx.x;
    const int rtile = blockIdx.x >> 3;                 // 8 n-groups per row tile
    const int ngrp  = blockIdx.x & 7;
    const int r0    = rtile * 16;

    for (int i = tid; i < PP; i += 128) ldsG[i] = g[i];
    for (int i = tid; i < 16 * LP; i += 128) {
        int rr = i >> 9, kk = i & (LP - 1);
        int rg = r0 + rr;
        ldsA[rr * 516 + kk] = (rg < R) ? sino[rg * LP + kk] : 0.0f;
    }
    __syncthreads();

    const int wave = tid >> 5;
    const int lane = tid & 31;
    const int n0   = (ngrp * 4 + wave) * 16;
    const int mn   = lane & 15;                        // A row / B col within tile
    const int kh   = lane >> 4;                        // K half select (0 or 1)

    v8f acc = {};
    for (int k0 = 0; k0 < LP; k0 += 4) {
        const int ka = k0 + kh * 2;
        v2f a, b;
        a.x = ldsA[mn * 516 + ka];
        a.y = ldsA[mn * 516 + ka + 1];
        b.x = ldsG[(n0 + mn - ka) & (PP - 1)];
        b.y = ldsG[(n0 + mn - ka - 1) & (PP - 1)];
        acc = __builtin_amdgcn_wmma_f32_16x16x4_f32(
            /*neg_a=*/false, a, /*neg_b=*/false, b,
            /*c_mod=*/(short)0, acc, /*reuse_a=*/false, /*reuse_b=*/false);
    }

    #pragma unroll
    for (int v = 0; v < 8; ++v) {
        int rg = r0 + v + kh * 8;
        if (rg < R) sf[rg * LP + n0 + mn] = acc[v];
    }
}

// ---- Kernel 4: backprojection ----------------------------------------------
// recon[b,i,j] = sum_a lerp(sf[b,a,:], (i-c)*cos(th_a) + (j-c)*sin(th_a) + (L-1)/2)
__global__ void __launch_bounds__(256) fbp_backproject(const float* __restrict__ sf,
                                                       const float* __restrict__ angles,
                                                       const int* __restrict__ circle,
                                                       float* __restrict__ out, int A) {
    __shared__ float cs[256];
    __shared__ float sn[256];
    const int tid = threadIdx.x;
    if (tid < A) {
        float th = angles[tid] + 1.57079632679489662f; // th = angle + pi/2
        cs[tid]  = cosf(th);
        sn[tid]  = sinf(th);
    }
    __syncthreads();

    const int idx = blockIdx.x * 256 + tid;
    const int j   = idx & (LP - 1);
    const int i   = (idx >> 9) & (LP - 1);
    const int b   = idx >> 18;

    const float dx = (float)(i - LP / 2);              // x runs along axis i
    const float dy = (float)(j - LP / 2);
    const float* base = sf + (size_t)b * A * LP;

    float acc = 0.0f;
    for (int a = 0; a < A; ++a) {
        float u  = fmaf(dx, cs[a], fmaf(dy, sn[a], (float)(LP - 1) * 0.5f));
        float fl = floorf(u);
        float fr = u - fl;
        int   i0 = (int)fl;
        int   i1 = i0 + 1;
        float w0 = (i0 >= 0 && i0 < LP) ? (1.0f - fr) : 0.0f;
        float w1 = (i1 >= 0 && i1 < LP) ? fr : 0.0f;
        int   c0 = i0 < 0 ? 0 : (i0 > LP - 1 ? LP - 1 : i0);
        int   c1 = i1 < 0 ? 0 : (i1 > LP - 1 ? LP - 1 : i1);
        const float* rowp = base + a * LP;
        acc = fmaf(w0, rowp[c0], acc);
        acc = fmaf(w1, rowp[c1], acc);
    }
    if (circle[0] != 0 && (dx * dx + dy * dy) > (float)((LP / 2) * (LP / 2)))
        acc = 0.0f;
    out[idx] = acc;
}

// ---------------------------------------------------------------------------
extern "C" void kernel_launch(void* const* d_in, const int* in_sizes, int n_in,
                              void* d_out, int out_size, void* d_ws, size_t ws_size,
                              hipStream_t stream) {
    const float* sino   = (const float*)d_in[0];
    const float* angles = (const float*)d_in[1];
    const int*   circle = (const int*)d_in[2];
    float*       out    = (float*)d_out;

    const int A = in_sizes[1];          // 181
    const int R = in_sizes[0] / LP;     // B*A rows = 362

    // workspace layout: [0, 8K) filt(f64) | [8K, 12K) g(f32) | [12K, ...) sf
    double* filt = (double*)d_ws;
    float*  g    = (float*)((char*)d_ws + 8192);
    float*  sf   = (float*)((char*)d_ws + 8192 + 4096);

    fbp_build_filt<<<PP / 256, 256, 0, stream>>>(filt);
    fbp_build_g<<<PP / 256, 256, 0, stream>>>(filt, g);

    const int RT = (R + 15) / 16;
    fbp_filter_gemm<<<RT * 8, 128, 0, stream>>>(sino, g, sf, R);

    fbp_backproject<<<out_size / 256, 256, 0, stream>>>(sf, angles, circle, out, A);
}